// LightGCN_16243566314052
// MI455X (gfx1250) — compile-verified
//
#include <hip/hip_runtime.h>
#include <hip/hip_bf16.h>
#include <stdint.h>

// LightGCN propagation for MI455X (gfx1250).
// Roofline: 768 MFLOP total vs ~3 GB of L2-resident traffic -> atomic/L2 bound.
// CDNA5 paths used:
//  - global_atomic_add_f32 (no-return, scope:SCOPE_DEV): native f32 RMW in L2,
//    no CAS loop. 64 atomics/edge is the irreducible cost of unsorted SpMM.
//  - global_load_async_to_lds_b32 + s_wait_asynccnt: edge stream staged into
//    LDS on the ASYNCcnt path (no VGPR return, LOADcnt left free for the
//    random b128 feature gathers); inner loop reads indices via ds broadcast.
//  - WMMA intentionally absent: density ~9e-8, unstructured -> no dense tiles.

#define N_LAYERS 3
#define TPB 256          // 8 wave32 waves per block
#define EPB 256          // edges staged per block (16 teams x 16 edges)

// Native L2 fp32 atomic-add, device scope (visible to subsequent kernels).
__device__ __forceinline__ void atomic_add_f32_dev(float* p, float v) {
#if defined(__gfx1250__)
    asm volatile("global_atomic_add_f32 %0, %1, off scope:SCOPE_DEV"
                 :
                 : "v"(p), "v"(v)
                 : "memory");
#else
    atomicAdd(p, v);
#endif
}

// x = concat(user,item); y = 0; out = 0.25 * x
__global__ void lgcn_init(const float* __restrict__ user,
                          const float* __restrict__ item,
                          float* __restrict__ x,
                          float* __restrict__ y,
                          float* __restrict__ out,
                          int u_elems, int n_elems) {
    int i4 = blockIdx.x * blockDim.x + threadIdx.x;   // float4 index
    int i  = i4 << 2;
    if (i >= n_elems) return;
    float4 v = (i < u_elems)
                   ? reinterpret_cast<const float4*>(user)[i4]
                   : reinterpret_cast<const float4*>(item)[(i - u_elems) >> 2];
    reinterpret_cast<float4*>(x)[i4] = v;
    reinterpret_cast<float4*>(y)[i4] = make_float4(0.f, 0.f, 0.f, 0.f);
    reinterpret_cast<float4*>(out)[i4] =
        make_float4(0.25f * v.x, 0.25f * v.y, 0.25f * v.z, 0.25f * v.w);
}

// Edge-parallel SpMM. Each block stages EPB edges (row/col/val) into LDS via
// the async-to-LDS path, then 16 teams of 16 lanes each process 16 edges:
// per edge one b128 gather of the source row chunk + 4 f32 L2 atomics.
__global__ void __launch_bounds__(TPB)
lgcn_spmm(const int* __restrict__ row,
          const int* __restrict__ col,
          const float* __restrict__ val,
          const float* __restrict__ x,
          float* __restrict__ y,
          int n_edges) {
    __shared__ int   s_row[EPB];
    __shared__ int   s_col[EPB];
    __shared__ float s_val[EPB];

    const int t    = threadIdx.x;
    const int base = blockIdx.x * EPB;
    int e = base + t;
    if (e >= n_edges) e = n_edges - 1;      // clamp: keep global addr valid

#if defined(__gfx1250__)
    {
        // Flat LDS address truncated to 32 bits == LDS byte offset (ISA 10.2).
        uint32_t ldr = (uint32_t)(uintptr_t)&s_row[t];
        uint32_t ldc = (uint32_t)(uintptr_t)&s_col[t];
        uint32_t ldv = (uint32_t)(uintptr_t)&s_val[t];
        uint32_t off = (uint32_t)e << 2;    // per-lane byte offset (GVS mode)
        asm volatile(
            "global_load_async_to_lds_b32 %0, %3, %4\n\t"
            "global_load_async_to_lds_b32 %1, %3, %5\n\t"
            "global_load_async_to_lds_b32 %2, %3, %6\n\t"
            "s_wait_asynccnt 0x0"
            :
            : "v"(ldr), "v"(ldc), "v"(ldv), "v"(off),
              "s"(row), "s"(col), "s"(val)
            : "memory");
    }
#else
    s_row[t] = row[e];
    s_col[t] = col[e];
    s_val[t] = val[e];
#endif
    __syncthreads();

    const int team = t >> 4;                // 0..15
    const int sub  = t & 15;                // float4 chunk within 64-wide row
    const int fo   = sub << 2;

    for (int k = 0; k < EPB / 16; ++k) {
        int le = (team << 4) + k;           // local edge id, contiguous per team
        int ge = base + le;
        if (ge >= n_edges) break;
        int   r = s_row[le];                // ds broadcast within the team
        int   c = s_col[le];
        float v = s_val[le];

        const float4 xv =
            *reinterpret_cast<const float4*>(x + ((size_t)c << 6) + fo);
        float* dst = y + ((size_t)r << 6) + fo;

        atomic_add_f32_dev(dst + 0, xv.x * v);
        atomic_add_f32_dev(dst + 1, xv.y * v);
        atomic_add_f32_dev(dst + 2, xv.z * v);
        atomic_add_f32_dev(dst + 3, xv.w * v);
    }
}

// out += 0.25 * y ; and zero the ping-pong buffer for the next layer.
__global__ void lgcn_accum_zero(float4* __restrict__ out,
                                const float4* __restrict__ y,
                                float4* __restrict__ xz, int n4) {
    int i = blockIdx.x * blockDim.x + threadIdx.x;
    if (i >= n4) return;
    float4 o = out[i];
    float4 a = y[i];
    o.x += 0.25f * a.x;
    o.y += 0.25f * a.y;
    o.z += 0.25f * a.z;
    o.w += 0.25f * a.w;
    out[i] = o;
    xz[i]  = make_float4(0.f, 0.f, 0.f, 0.f);
}

extern "C" void kernel_launch(void* const* d_in, const int* in_sizes, int n_in,
                              void* d_out, int out_size, void* d_ws, size_t ws_size,
                              hipStream_t stream) {
    const float* user = (const float*)d_in[0];   // U*64 f32
    const float* item = (const float*)d_in[1];   // I*64 f32
    const int*   row  = (const int*)d_in[2];     // 2*NNZ i32
    const int*   col  = (const int*)d_in[3];     // 2*NNZ i32
    const float* val  = (const float*)d_in[4];   // 2*NNZ f32

    const int u_elems = in_sizes[0];
    const int n_elems = in_sizes[0] + in_sizes[1];   // N*64
    const int n_edges = in_sizes[2];
    const int n4      = n_elems >> 2;

    float* x   = (float*)d_ws;               // ping
    float* y   = x + (size_t)n_elems;        // pong (2*N*64*4 = 76.8 MB total)
    float* out = (float*)d_out;

    const int grid_n4 = (n4 + TPB - 1) / TPB;
    const int grid_e  = (n_edges + EPB - 1) / EPB;

    lgcn_init<<<grid_n4, TPB, 0, stream>>>(user, item, x, y, out,
                                           u_elems, n_elems);

    for (int l = 0; l < N_LAYERS; ++l) {
        lgcn_spmm<<<grid_e, TPB, 0, stream>>>(row, col, val, x, y, n_edges);
        // out += 0.25*y; zero old x so it can serve as next layer's y.
        lgcn_accum_zero<<<grid_n4, TPB, 0, stream>>>(
            (float4*)out, (const float4*)y, (float4*)x, n4);
        float* tmp = x; x = y; y = tmp;      // layer output becomes next input
    }
}